// lupiEncoder_84713934946331
// MI455X (gfx1250) — compile-verified
//
#include <hip/hip_runtime.h>
#include <hip/hip_bf16.h>

// Problem constants (from reference)
#define N_GRAPHS 32
#define NODESG   4096                 // nodes per graph
#define D_IN     128
#define HID      16
#define OUT_DIM  39
#define NTOT     (N_GRAPHS * NODESG)  // 131072
#define EPG      (NTOT * 32 / N_GRAPHS) // edges per graph = 131072
#define GSIZE    (NODESG * HID)       // 65536 floats per graph of h_flat

typedef __attribute__((ext_vector_type(2))) float v2f;
typedef __attribute__((ext_vector_type(8))) float v8f;

// ---------------------------------------------------------------------------
// Kernel 1: fused xl = x@W_l^T  and  xrb = x@W_r^T + b_l  via WMMA f32 16x16x4.
// One wave per 16-node tile; K-loop of 32 steps covers D_IN=128.
// A lane layout (ISA 16x4 f32 A): lane<16 -> M=lane, K pair {4t,4t+1};
//                                 lane>=16 -> M=lane-16, K pair {4t+2,4t+3}.
// B from W (row-major [HID, D_IN]): B[k][n] = W[n][k] -> contiguous float2.
// C/D: VGPR r holds M = r + 8*(lane>=16), N = lane&15.
// ---------------------------------------------------------------------------
__global__ __launch_bounds__(128)
void lupi_lin_kernel(const float* __restrict__ x,
                     const float* __restrict__ Wl,
                     const float* __restrict__ bl,
                     const float* __restrict__ Wr,
                     float* __restrict__ xl,
                     float* __restrict__ xrb)
{
    const int lane = threadIdx.x & 31;
    const int wave = threadIdx.x >> 5;
    const int tile = blockIdx.x * 4 + wave;   // 16-node tile index
    const int m    = lane & 15;               // A-matrix row (M)
    const int hi   = lane >> 4;               // K-half select
    const int n    = lane & 15;               // B-matrix col (N) == HID index

    const float* xrow  = x  + (size_t)(tile * 16 + m) * D_IN + 2 * hi;
    const float* wlrow = Wl + (size_t)n * D_IN + 2 * hi;
    const float* wrrow = Wr + (size_t)n * D_IN + 2 * hi;

    v8f cl = {};
    v8f cr = {};
#pragma unroll 8
    for (int t = 0; t < D_IN / 4; ++t) {
        v2f a  = __builtin_nontemporal_load((const v2f*)(xrow + 4 * t)); // x: streamed once
        v2f bL = *(const v2f*)(wlrow + 4 * t);                           // weights: cache-hot
        v2f bR = *(const v2f*)(wrrow + 4 * t);
        cl = __builtin_amdgcn_wmma_f32_16x16x4_f32(false, a, false, bL,
                                                   (short)0, cl, false, false);
        cr = __builtin_amdgcn_wmma_f32_16x16x4_f32(false, a, false, bR,
                                                   (short)0, cr, false, false);
    }

    const float bias = bl[n];
#pragma unroll
    for (int r = 0; r < 8; ++r) {
        const size_t node = (size_t)tile * 16 + r + 8 * hi;
        xl [node * HID + n] = cl[r];          // lanes 0-15 / 16-31 -> two 64B segments
        xrb[node * HID + n] = cr[r] + bias;
    }
}

// ---------------------------------------------------------------------------
// Kernel 2: one workgroup per graph (256KB agg + 16KB cnt live in LDS).
//  Phase A: stream this graph's 131072 edges (stride-32 in global edge id),
//           gather xl[src] (coalesced 64B per 16 lanes), ds_add_f32 into LDS.
//           Each edge's 16 adds hit 16 consecutive banks; 2 edges per wave.
//  Phase B: h = relu(agg/max(cnt,1) + xrb), fused dot with W_out (L2-resident),
//           39 per-thread accumulators, deterministic shuffle+LDS reduction.
// ---------------------------------------------------------------------------
__global__ __launch_bounds__(1024)
void lupi_graph_kernel(const int*   __restrict__ ei,    // [2, E] int32
                       const float* __restrict__ xl,
                       const float* __restrict__ xrb,
                       const float* __restrict__ Wout,  // [OUT_DIM, GSIZE]
                       const float* __restrict__ bout,
                       float*       __restrict__ out)   // [N_GRAPHS, OUT_DIM]
{
    extern __shared__ float smem[];        // GSIZE agg + NODESG cnt = 278528 B
    float* agg = smem;
    float* cnt = smem + GSIZE;

    const int g    = blockIdx.x;
    const int tid  = threadIdx.x;
    const int lane = tid & 31;
    const int wave = tid >> 5;

    for (int i = tid; i < GSIZE + NODESG; i += 1024) smem[i] = 0.0f;
    __syncthreads();

    // ---- Phase A: edge scatter into LDS ----
    const int h    = lane & 15;            // HID channel handled by this lane
    const int slot = lane >> 4;            // which of 2 edges in this wave step
    const long long Etot = (long long)N_GRAPHS * EPG;

    for (int it = 0; it < EPG / 64; ++it) {
        const int j = it * 64 + wave * 2 + slot;          // edge j of graph g
        const long long e = (long long)j * N_GRAPHS + g;  // global edge id
        const int src = __builtin_nontemporal_load(ei + e);        // streamed once
        const int dst = __builtin_nontemporal_load(ei + Etot + e);
        const float v = xl[(size_t)src * HID + h];        // 64B coalesced gather
        atomicAdd(&agg[(dst & (NODESG - 1)) * HID + h], v);   // ds_add_f32
        if (h == 0) atomicAdd(&cnt[dst & (NODESG - 1)], 1.0f);
    }
    __syncthreads();

    // ---- Phase B: mean + bias + relu, fused W_out reduction ----
    float acc[OUT_DIM];
#pragma unroll
    for (int o = 0; o < OUT_DIM; ++o) acc[o] = 0.0f;

    for (int f = tid; f < GSIZE; f += 1024) {
        const float c    = cnt[f >> 4];
        const float mean = agg[f] / fmaxf(c, 1.0f);
        float hv = mean + xrb[(size_t)g * GSIZE + f];     // xrb already has +b_l
        hv = fmaxf(hv, 0.0f);
#pragma unroll
        for (int o = 0; o < OUT_DIM; ++o)                 // Wout coalesced per o
            acc[o] = fmaf(hv, Wout[(size_t)o * GSIZE + f], acc[o]);
    }
    __syncthreads();   // everyone done reading agg -> smem reusable

    // wave32 shuffle reduction, then fixed-order cross-wave sum (deterministic)
#pragma unroll
    for (int o = 0; o < OUT_DIM; ++o) {
        float v = acc[o];
        for (int off = 16; off > 0; off >>= 1)
            v += __shfl_down(v, off, 32);
        if (lane == 0) smem[wave * OUT_DIM + o] = v;
    }
    __syncthreads();

    if (tid < OUT_DIM) {
        float s = bout[tid];
        for (int w = 0; w < 32; ++w) s += smem[w * OUT_DIM + tid];
        out[g * OUT_DIM + tid] = s;   // output shape [G, OUT_DIM, 1]
    }
}

// ---------------------------------------------------------------------------
extern "C" void kernel_launch(void* const* d_in, const int* in_sizes, int n_in,
                              void* d_out, int out_size, void* d_ws, size_t ws_size,
                              hipStream_t stream)
{
    const float* x    = (const float*)d_in[0];
    const int*   ei   = (const int*)  d_in[1];
    const float* Wl   = (const float*)d_in[2];
    const float* bl   = (const float*)d_in[3];
    const float* Wr   = (const float*)d_in[4];
    const float* Wout = (const float*)d_in[5];
    const float* bout = (const float*)d_in[6];
    float*       out  = (float*)d_out;

    float* xl  = (float*)d_ws;                 //  8 MB
    float* xrb = xl + (size_t)NTOT * HID;      // +8 MB  (16 MB total scratch)

    // K1: 8192 tiles of 16 nodes, 4 waves per 128-thread block -> 2048 blocks
    lupi_lin_kernel<<<2048, 128, 0, stream>>>(x, Wl, bl, Wr, xl, xrb);

    // K2: one workgroup (= one WGP, 272KB LDS) per graph
    const size_t lds_bytes = (size_t)(GSIZE + NODESG) * sizeof(float);
    lupi_graph_kernel<<<N_GRAPHS, 1024, lds_bytes, stream>>>(ei, xl, xrb,
                                                             Wout, bout, out);
}